// JointNetwork_84404697301336
// MI455X (gfx1250) — compile-verified
//
#include <hip/hip_runtime.h>

typedef float v2f __attribute__((ext_vector_type(2)));
typedef float v4f __attribute__((ext_vector_type(4)));
typedef float v8f __attribute__((ext_vector_type(8)));

#define K_DIM   512     // ENC_DIM == PRED_DIM
#define W_LD    1024    // W row stride (ENC_DIM + PRED_DIM)
#define VOCAB   2048
#define T_LEN   256
#define U_LEN   64

// ---------------------------------------------------------------------------
// Phase 1: C[m, n] = sum_k A[m, k] * Wbase[n*W_LD + k]  (+ bias[n] if given)
// One wave computes a 16x16 fp32 tile via V_WMMA_F32_16X16X4_F32.
// Block = 256 threads = 8 waves -> 16 (M) x 128 (N) per block.
// grid.x = M/16, grid.y = N/128
// ---------------------------------------------------------------------------
__global__ __launch_bounds__(256) void wmma_proj_kernel(
    const float* __restrict__ A,      // M x K_DIM, row-major
    const float* __restrict__ Wbase,  // row n at Wbase + n*W_LD (col offset pre-applied)
    const float* __restrict__ bias,   // VOCAB or nullptr
    float* __restrict__ C)            // M x VOCAB, row-major
{
  const int lane  = threadIdx.x & 31;
  const int wave  = threadIdx.x >> 5;
  const int half  = lane >> 4;   // 0: lanes 0-15, 1: lanes 16-31
  const int l15   = lane & 15;
  const int mBase = blockIdx.x * 16;
  const int nBase = blockIdx.y * 128 + wave * 16;

  // A 16x4 f32 frag: lane l15 -> row m; lane-half -> K pair; VGPR -> K within pair
  const float* Arow = A + (size_t)(mBase + l15) * K_DIM + half * 2;
  // B 4x16 f32 frag: lane l15 -> col n; same K striping as A
  const float* Wrow = Wbase + (size_t)(nBase + l15) * W_LD + half * 2;

  v8f acc = {};
#pragma unroll 4
  for (int k = 0; k < K_DIM; k += 4) {
    v2f afrag = *(const v2f*)(Arow + k);
    v2f bfrag = *(const v2f*)(Wrow + k);
    // 8 args: (neg_a, A, neg_b, B, c_mod, C, reuse_a, reuse_b)
    acc = __builtin_amdgcn_wmma_f32_16x16x4_f32(
        false, afrag, false, bfrag, (short)0, acc, false, false);
  }

  const float bv = bias ? bias[nBase + l15] : 0.0f;
  // C layout: VGPR i -> rows (mBase + i) [lanes 0-15] and (mBase + i + 8) [lanes 16-31]
  float* Crow = C + (size_t)(mBase + half * 8) * VOCAB + nBase + l15;
#pragma unroll
  for (int i = 0; i < 8; ++i)
    Crow[(size_t)i * VOCAB] = acc[i] + bv;
}

// ---------------------------------------------------------------------------
// Phase 2: out[r, u, v] = encP[r, v] + pb[b, u, v]   (bias already in pb)
// r = b*T + t.  One block per r; enc row (8 KB) cached in registers; 64 u-iters
// of 128-bit non-temporal stores (512 MB write-once stream; keep L2 for pb).
// ---------------------------------------------------------------------------
__global__ __launch_bounds__(256) void bcast_add_kernel(
    const float* __restrict__ encP,   // (B*T) x VOCAB
    const float* __restrict__ pb,     // (B*U) x VOCAB  (pred_proj + bias)
    float* __restrict__ out)          // (B*T*U) x VOCAB
{
  const int r   = blockIdx.x;     // 0 .. B*T-1
  const int b   = r >> 8;         // T_LEN == 256
  const int tid = threadIdx.x;

  const v4f* encRow = (const v4f*)(encP + (size_t)r * VOCAB);
  const v4f e0 = encRow[tid];
  const v4f e1 = encRow[tid + 256];

  const v4f* pbB  = (const v4f*)(pb + (size_t)b * U_LEN * VOCAB);
  v4f*       outB = (v4f*)(out + (size_t)r * U_LEN * VOCAB);

#pragma unroll 4
  for (int u = 0; u < U_LEN; ++u) {
    const v4f* pr = pbB + (size_t)u * (VOCAB / 4);
    v4f*     orow = outB + (size_t)u * (VOCAB / 4);
    v4f p0 = pr[tid];
    v4f p1 = pr[tid + 256];
    __builtin_nontemporal_store(e0 + p0, orow + tid);
    __builtin_nontemporal_store(e1 + p1, orow + tid + 256);
  }
}

extern "C" void kernel_launch(void* const* d_in, const int* in_sizes, int n_in,
                              void* d_out, int out_size, void* d_ws, size_t ws_size,
                              hipStream_t stream) {
  const float* enc  = (const float*)d_in[0];  // (4, 256, 512)
  const float* pred = (const float*)d_in[1];  // (4, 64, 512)
  const float* W    = (const float*)d_in[2];  // (2048, 1024)
  const float* bias = (const float*)d_in[3];  // (2048,)
  float* out = (float*)d_out;                 // (4, 256, 64, 2048)

  float* encP = (float*)d_ws;                 // 1024 x 2048 fp32 = 8 MB
  float* pb   = encP + (size_t)1024 * VOCAB;  //  256 x 2048 fp32 = 2 MB

  // enc_proj: M=1024, N=2048, K=512, W cols [0,512)
  wmma_proj_kernel<<<dim3(1024 / 16, VOCAB / 128), 256, 0, stream>>>(
      enc, W, nullptr, encP);

  // pred_proj + bias: M=256, N=2048, K=512, W cols [512,1024)
  wmma_proj_kernel<<<dim3(256 / 16, VOCAB / 128), 256, 0, stream>>>(
      pred, W + K_DIM, bias, pb);

  // broadcast add: 512 MB of NT stores — the bandwidth-bound phase
  bcast_add_kernel<<<dim3(1024), 256, 0, stream>>>(encP, pb, out);
}